// MambaBlock_15822659518963
// MI455X (gfx1250) — compile-verified
//
#include <hip/hip_runtime.h>
#include <hip/hip_bf16.h>
#include <math.h>

#define D_MODEL 1024
#define D_STATE 16
#define D_CONV  4
#define D_INNER 2048
#define BB      4
#define LL      2048
#define NTOK    (BB * LL)        // 8192

typedef __attribute__((ext_vector_type(2))) float v2f;
typedef __attribute__((ext_vector_type(8))) float v8f;

// ---------------------------------------------------------------------------
// Kernel 1: LayerNorm. One 256-thread block per row of 1024.
// ---------------------------------------------------------------------------
__global__ void ln_kernel(const float* __restrict__ x,
                          const float* __restrict__ w,
                          const float* __restrict__ b,
                          float* __restrict__ xn) {
    const int tid  = threadIdx.x;
    const int lane = tid & 31;
    const int wv   = tid >> 5;
    const size_t base = (size_t)blockIdx.x * D_MODEL;

    float v[4];
    float sum = 0.f, sq = 0.f;
#pragma unroll
    for (int i = 0; i < 4; ++i) {
        v[i] = x[base + tid + i * 256];
        sum += v[i];
        sq  += v[i] * v[i];
    }
#pragma unroll
    for (int off = 16; off >= 1; off >>= 1) {
        sum += __shfl_xor(sum, off, 32);
        sq  += __shfl_xor(sq,  off, 32);
    }
    __shared__ float s1[8], s2[8];
    if (lane == 0) { s1[wv] = sum; s2[wv] = sq; }
    __syncthreads();
    float tot = 0.f, totq = 0.f;
#pragma unroll
    for (int j = 0; j < 8; ++j) { tot += s1[j]; totq += s2[j]; }
    const float mu  = tot * (1.f / D_MODEL);
    const float var = totq * (1.f / D_MODEL) - mu * mu;
    const float inv = rsqrtf(var + 1e-5f);
#pragma unroll
    for (int i = 0; i < 4; ++i) {
        const int c = tid + i * 256;
        xn[base + c] = (v[i] - mu) * inv * w[c] + b[c];
    }
}

// ---------------------------------------------------------------------------
// Kernel 2/5: fp32 WMMA GEMM  C[M,N] = A[M,K] @ B[K,N] (+ residual)
// Block = 256 threads (8 waves, 2x4 wave grid). Block tile 64 x 128.
// Each wave owns a 2x2 grid of 16x16 accumulators -> each A/B fragment
// feeds two v_wmma_f32_16x16x4_f32 (2 LDS dwords/lane/WMMA).
// K staged through LDS in chunks of 32; next chunk prefetched from global.
// ---------------------------------------------------------------------------
#define KC        32
#define SA_STRIDE 36    // 64 x 32 A tile, padded
#define SB_STRIDE 136   // 32 x 128 B tile, padded

__global__ void gemm_wmma_f32(const float* __restrict__ A,
                              const float* __restrict__ Bw,
                              float* __restrict__ C,
                              int M, int N, int K,
                              const float* __restrict__ res) {
    __shared__ float sA[64 * SA_STRIDE];
    __shared__ float sB[KC * SB_STRIDE];

    const int tid    = threadIdx.x;
    const int lane   = tid & 31;
    const int wv     = tid >> 5;
    const int waveM  = wv & 1;          // 0..1
    const int waveN  = wv >> 1;         // 0..3
    const int laneLo = lane & 15;
    const int half   = lane >> 4;       // 0 or 1

    const int m0  = blockIdx.x * 64;
    const int bn0 = blockIdx.y * 128;

    v8f acc00 = {}, acc01 = {}, acc10 = {}, acc11 = {};

    // per-thread staging coordinates
    const int rA  = tid >> 3;           // 0..31? -> 64 rows via 2 passes below
    const int c4A = tid & 7;            // 8 float4 per 32-wide A row
    const int rB  = tid >> 5;           // with 4 passes -> 32 rows
    const int c4B = tid & 31;           // 32 float4 per 128-wide B row

    for (int k0 = 0; k0 < K; k0 += KC) {
        __syncthreads();
        // --- stage A tile: 64 rows x 32 cols (2 float4 per thread) ---
#pragma unroll
        for (int t = 0; t < 2; ++t) {
            const int r = rA + t * 32;  // 0..63
            const float4 va =
                *(const float4*)&A[(size_t)(m0 + r) * K + k0 + c4A * 4];
            *(float4*)&sA[r * SA_STRIDE + c4A * 4] = va;
        }
        // --- stage B tile: 32 rows x 128 cols (4 float4 per thread) ---
#pragma unroll
        for (int t = 0; t < 4; ++t) {
            const int r = rB + t * 8;   // 0..31
            const float4 vb =
                *(const float4*)&Bw[(size_t)(k0 + r) * N + bn0 + c4B * 4];
            *(float4*)&sB[r * SB_STRIDE + c4B * 4] = vb;
        }
        // --- prefetch next chunk while this one computes ---
        if (k0 + KC < K) {
            __builtin_prefetch(&A[(size_t)(m0 + rA) * K + k0 + KC + c4A * 4], 0, 1);
            __builtin_prefetch(&Bw[(size_t)(k0 + KC + rB) * N + bn0 + c4B * 4], 0, 1);
        }
        __syncthreads();

#pragma unroll
        for (int kk = 0; kk < KC; kk += 4) {
            const int ko = kk + half * 2;
            v2f a0, a1, b0, b1;
            // A frags (16x4): lanes 0-15 K=ko,ko+1 ; lanes 16-31 K=ko+2,ko+3
            const int ar0 = (waveM * 32 + laneLo) * SA_STRIDE + ko;
            a0.x = sA[ar0];
            a0.y = sA[ar0 + 1];
            const int ar1 = ar0 + 16 * SA_STRIDE;
            a1.x = sA[ar1];
            a1.y = sA[ar1 + 1];
            // B frags (4x16): N across lanes, K split across lane halves
            const int bc0 = ko * SB_STRIDE + waveN * 32 + laneLo;
            b0.x = sB[bc0];
            b0.y = sB[bc0 + SB_STRIDE];
            const int bc1 = bc0 + 16;
            b1.x = sB[bc1];
            b1.y = sB[bc1 + SB_STRIDE];
            acc00 = __builtin_amdgcn_wmma_f32_16x16x4_f32(
                false, a0, false, b0, (short)0, acc00, false, false);
            acc01 = __builtin_amdgcn_wmma_f32_16x16x4_f32(
                false, a0, false, b1, (short)0, acc01, false, false);
            acc10 = __builtin_amdgcn_wmma_f32_16x16x4_f32(
                false, a1, false, b0, (short)0, acc10, false, false);
            acc11 = __builtin_amdgcn_wmma_f32_16x16x4_f32(
                false, a1, false, b1, (short)0, acc11, false, false);
        }
    }

    // --- store: VGPR i -> row (i + half*8), col = lane&15 within each tile ---
#pragma unroll
    for (int tM = 0; tM < 2; ++tM) {
#pragma unroll
        for (int tN = 0; tN < 2; ++tN) {
            const v8f acc = tM == 0 ? (tN == 0 ? acc00 : acc01)
                                    : (tN == 0 ? acc10 : acc11);
            const int col = bn0 + waveN * 32 + tN * 16 + laneLo;
#pragma unroll
            for (int i = 0; i < 8; ++i) {
                const int row = m0 + waveM * 32 + tM * 16 + i + half * 8;
                float v = acc[i];
                if (res) v += res[(size_t)row * N + col];
                C[(size_t)row * N + col] = v;
            }
        }
    }
}

// ---------------------------------------------------------------------------
// Kernel 3: causal depthwise conv4 + bias + SiLU, then x_proj (2048 -> 33).
// One block per token (b,l); 256 threads x 8 channels each.
// ---------------------------------------------------------------------------
__global__ void conv_xproj_kernel(const float* __restrict__ xz,
                                  const float* __restrict__ conv_w,
                                  const float* __restrict__ conv_b,
                                  const float* __restrict__ xproj,
                                  float* __restrict__ xc,
                                  float* __restrict__ dbc0,
                                  float* __restrict__ Bp,
                                  float* __restrict__ Cp) {
    const int tok = blockIdx.x;
    const int bidx = tok / LL;
    const int l    = tok % LL;
    const int tid  = threadIdx.x;
    const int lane = tid & 31;
    const int wv   = tid >> 5;

    float acc[33];
#pragma unroll
    for (int j = 0; j < 33; ++j) acc[j] = 0.f;

#pragma unroll
    for (int t = 0; t < 8; ++t) {
        const int d = tid + t * 256;
        float s = conv_b[d];
#pragma unroll
        for (int k = 0; k < D_CONV; ++k) {
            const int ll = l - (D_CONV - 1) + k;
            if (ll >= 0)
                s += xz[((size_t)bidx * LL + ll) * (2 * D_INNER) + d] *
                     conv_w[d * D_CONV + k];
        }
        s = s / (1.f + __expf(-s));           // SiLU
        xc[(size_t)tok * D_INNER + d] = s;
#pragma unroll
        for (int j = 0; j < 33; ++j)
            acc[j] += s * xproj[(size_t)d * 33 + j];
    }

    // wave reduction of 33 partial dots
#pragma unroll
    for (int j = 0; j < 33; ++j) {
#pragma unroll
        for (int off = 16; off >= 1; off >>= 1)
            acc[j] += __shfl_xor(acc[j], off, 32);
    }
    __shared__ float sred[8 * 33];
    if (lane == 0) {
#pragma unroll
        for (int j = 0; j < 33; ++j) sred[wv * 33 + j] = acc[j];
    }
    __syncthreads();
    if (tid < 33) {
        float v = 0.f;
#pragma unroll
        for (int w8 = 0; w8 < 8; ++w8) v += sred[w8 * 33 + tid];
        if (tid == 0)       dbc0[tok] = v;
        else if (tid < 17)  Bp[(size_t)tok * D_STATE + (tid - 1)]  = v;
        else                Cp[(size_t)tok * D_STATE + (tid - 17)] = v;
    }
}

// ---------------------------------------------------------------------------
// Kernel 4: selective scan over L, fused with +xc*D and *silu(z).
// grid = (D_INNER/256, B); each thread owns one (b,d) channel, h[16] in VGPRs.
// ---------------------------------------------------------------------------
__global__ void scan_kernel(const float* __restrict__ xc,
                            const float* __restrict__ xz,
                            const float* __restrict__ dbc0,
                            const float* __restrict__ Bp,
                            const float* __restrict__ Cp,
                            const float* __restrict__ dt_w,
                            const float* __restrict__ dt_b,
                            const float* __restrict__ A_log,
                            const float* __restrict__ Dp,
                            float* __restrict__ yact) {
    const int tid  = threadIdx.x;
    const int d    = blockIdx.x * 256 + tid;
    const int bidx = blockIdx.y;

    float Ai[D_STATE], h[D_STATE];
#pragma unroll
    for (int n = 0; n < D_STATE; ++n) {
        Ai[n] = -__expf(A_log[(size_t)d * D_STATE + n]);
        h[n]  = 0.f;
    }
    const float dtw = dt_w[d];
    const float dtb = dt_b[d];
    const float Dd  = Dp[d];

    __shared__ float sB[D_STATE];
    __shared__ float sC[D_STATE];
    __shared__ float s0;

    for (int l = 0; l < LL; ++l) {
        const size_t tok = (size_t)bidx * LL + l;
        if (tid < 16)       sB[tid]      = Bp[tok * D_STATE + tid];
        else if (tid < 32)  sC[tid - 16] = Cp[tok * D_STATE + (tid - 16)];
        else if (tid == 32) s0           = dbc0[tok];
        __syncthreads();

        const float x_t = xc[tok * D_INNER + d];
        const float zv  = xz[tok * (2 * D_INNER) + D_INNER + d];
        const float u   = s0 * dtw + dtb;
        const float dt  = (u > 20.f) ? u : log1pf(__expf(u));

        float y = 0.f;
#pragma unroll
        for (int n = 0; n < D_STATE; ++n) {
            const float ab = __expf(dt * Ai[n]);
            h[n] = ab * h[n] + (dt * sB[n]) * x_t;
            y += h[n] * sC[n];
        }
        y = (y + x_t * Dd) * (zv / (1.f + __expf(-zv)));   // * silu(z)
        yact[tok * D_INNER + d] = y;
        __syncthreads();
    }
}

// ---------------------------------------------------------------------------
extern "C" void kernel_launch(void* const* d_in, const int* in_sizes, int n_in,
                              void* d_out, int out_size, void* d_ws, size_t ws_size,
                              hipStream_t stream) {
    const float* x       = (const float*)d_in[0];
    const float* norm_w  = (const float*)d_in[1];
    const float* norm_b  = (const float*)d_in[2];
    const float* W_in    = (const float*)d_in[3];
    const float* conv_w  = (const float*)d_in[4];
    const float* conv_b  = (const float*)d_in[5];
    const float* x_projw = (const float*)d_in[6];
    const float* dt_w    = (const float*)d_in[7];
    const float* dt_b    = (const float*)d_in[8];
    const float* A_log   = (const float*)d_in[9];
    const float* Dp      = (const float*)d_in[10];
    const float* W_out   = (const float*)d_in[11];
    float* out = (float*)d_out;

    // workspace layout (floats)
    float* ws   = (float*)d_ws;
    float* xn   = ws;                                   // 8192*1024
    float* xz   = xn   + (size_t)NTOK * D_MODEL;        // 8192*4096
    float* xc   = xz   + (size_t)NTOK * 2 * D_INNER;    // 8192*2048
    float* dbc0 = xc   + (size_t)NTOK * D_INNER;        // 8192
    float* Bp   = dbc0 + NTOK;                          // 8192*16
    float* Cp   = Bp   + (size_t)NTOK * D_STATE;        // 8192*16
    float* yact = Cp   + (size_t)NTOK * D_STATE;        // 8192*2048

    // 1) LayerNorm
    ln_kernel<<<NTOK, 256, 0, stream>>>(x, norm_w, norm_b, xn);

    // 2) xz = xn @ W_in   (8192x1024 @ 1024x4096)  -- WMMA fp32
    gemm_wmma_f32<<<dim3(NTOK / 64, (2 * D_INNER) / 128), 256, 0, stream>>>(
        xn, W_in, xz, NTOK, 2 * D_INNER, D_MODEL, nullptr);

    // 3) conv + SiLU + x_proj
    conv_xproj_kernel<<<NTOK, 256, 0, stream>>>(
        xz, conv_w, conv_b, x_projw, xc, dbc0, Bp, Cp);

    // 4) selective scan fused with +xc*D and *silu(z)
    scan_kernel<<<dim3(D_INNER / 256, BB), 256, 0, stream>>>(
        xc, xz, dbc0, Bp, Cp, dt_w, dt_b, A_log, Dp, yact);

    // 5) out = yact @ W_out + residual x  (8192x2048 @ 2048x1024) -- WMMA fp32
    gemm_wmma_f32<<<dim3(NTOK / 64, D_MODEL / 128), 256, 0, stream>>>(
        yact, W_out, out, NTOK, D_MODEL, D_INNER, x);
}